// GCN_43533788512795
// MI455X (gfx1250) — compile-verified
//
#include <hip/hip_runtime.h>

typedef __attribute__((ext_vector_type(2))) float v2f;
typedef __attribute__((ext_vector_type(8))) float v8f;

#define N_NODES  50000
#define E_DIM    128
#define H_DIM    256
#define OUT_DIM  2
#define N_EDGES  800000
#define NEG_SLOPE 0.01f

// ---------------- degree / norm ----------------
__global__ void k_deg_init(float* __restrict__ deg) {
  int i = blockIdx.x * blockDim.x + threadIdx.x;
  if (i < N_NODES) deg[i] = 1.0f;   // self-loop
}

__global__ void k_deg_count(const int* __restrict__ dst, float* __restrict__ deg) {
  int e = blockIdx.x * blockDim.x + threadIdx.x;
  if (e < N_EDGES) atomicAdd(&deg[dst[e]], 1.0f);
}

__global__ void k_dis(const float* __restrict__ deg, float* __restrict__ dis) {
  int i = blockIdx.x * blockDim.x + threadIdx.x;
  if (i < N_NODES) dis[i] = rsqrtf(deg[i]);   // deg >= 1 always
}

// ---------------- WMMA fp32 GEMM: out = [act](A @ W [+ bias]) ----------------
// A: [N_NODES, K] row-major (optionally A - A2 elementwise), W: [K, H_DIM] row-major.
// One wave per 16x16 tile of the output. V_WMMA_F32_16X16X4_F32 per 4-wide K step.
template<int K, bool ACT, bool DIFF>
__global__ void k_gemm_wmma(const float* __restrict__ A, const float* __restrict__ A2,
                            const float* __restrict__ W, const float* __restrict__ bias,
                            float* __restrict__ out) {
  const int wavesPerBlock = blockDim.x >> 5;
  const int wave = blockIdx.x * wavesPerBlock + (threadIdx.x >> 5);
  const int lane = threadIdx.x & 31;
  const int lo = lane & 15;       // m (A) / n (B) index within tile
  const int hi = lane >> 4;       // which half of the K quad
  const int ntilesN = H_DIM / 16; // 16
  const int m0 = (wave / ntilesN) * 16;
  const int n0 = (wave % ntilesN) * 16;

  const float* __restrict__ arow  = A  + (size_t)(m0 + lo) * K;
  const float* __restrict__ arow2 = DIFF ? (A2 + (size_t)(m0 + lo) * K) : nullptr;

  v8f c = {};
#pragma unroll 8
  for (int k0 = 0; k0 < K; k0 += 4) {
    const int ka = k0 + 2 * hi;
    v2f a, b;
    if (DIFF) {
      a.x = arow[ka]     - arow2[ka];
      a.y = arow[ka + 1] - arow2[ka + 1];
    } else {
      a.x = arow[ka];
      a.y = arow[ka + 1];
    }
    b.x = W[(size_t)ka       * H_DIM + n0 + lo];
    b.y = W[(size_t)(ka + 1) * H_DIM + n0 + lo];
    c = __builtin_amdgcn_wmma_f32_16x16x4_f32(false, a, false, b, (short)0, c,
                                              false, false);
  }

#pragma unroll
  for (int r = 0; r < 8; ++r) {
    const int m = m0 + r + 8 * hi;   // C/D layout: vgpr r -> rows r and r+8
    const int n = n0 + lo;
    float v = c[r];
    if (ACT) {
      v += bias[n];
      v = v > 0.0f ? v : NEG_SLOPE * v;
    }
    out[(size_t)m * H_DIM + n] = v;
  }
}

// ---------------- self-loop init: out = h * dis^2 + bias ----------------
__global__ void k_selfinit(const float* __restrict__ h, const float* __restrict__ dis,
                           const float* __restrict__ bias, float* __restrict__ out) {
  int t = blockIdx.x * blockDim.x + threadIdx.x;
  if (t < N_NODES * H_DIM) {
    int i = t >> 8;           // / H_DIM
    int cc = t & (H_DIM - 1);
    float di = dis[i];
    out[t] = h[t] * (di * di) + bias[cc];
  }
}

// ---------------- edge scatter: out[dst] += h[src] * dis[s]*dis[d] ----------------
// one thread = one edge x 4 channels (float4 gather, 4 f32 atomics)
__global__ void k_scatter(const int* __restrict__ src, const int* __restrict__ dst,
                          const float* __restrict__ dis, const float* __restrict__ h,
                          float* __restrict__ out) {
  int t = blockIdx.x * blockDim.x + threadIdx.x;   // < N_EDGES * 64
  int e = t >> 6;
  if (e < N_EDGES) {
    int c4 = (t & 63) << 2;
    int s = src[e], d = dst[e];
    float norm = dis[s] * dis[d];
    const float4 hv = *reinterpret_cast<const float4*>(h + (size_t)s * H_DIM + c4);
    float* o = out + (size_t)d * H_DIM + c4;
    atomicAdd(o + 0, hv.x * norm);
    atomicAdd(o + 1, hv.y * norm);
    atomicAdd(o + 2, hv.z * norm);
    atomicAdd(o + 3, hv.w * norm);
  }
}

// ---------------- skinny projection: out[i,j] = X[i,:] @ Wo[:,j] + bo[j] ----------------
__global__ void k_outproj(const float* __restrict__ X, const float* __restrict__ Wo,
                          const float* __restrict__ bo, float* __restrict__ out) {
  int t = blockIdx.x * blockDim.x + threadIdx.x;   // < N_NODES * OUT_DIM
  if (t < N_NODES * OUT_DIM) {
    int i = t >> 1;
    int j = t & 1;
    const float4* row = reinterpret_cast<const float4*>(X + (size_t)i * H_DIM);
    float acc = bo[j];
#pragma unroll 8
    for (int k = 0; k < H_DIM / 4; ++k) {
      float4 x = row[k];
      acc += x.x * Wo[(4 * k + 0) * OUT_DIM + j];
      acc += x.y * Wo[(4 * k + 1) * OUT_DIM + j];
      acc += x.z * Wo[(4 * k + 2) * OUT_DIM + j];
      acc += x.w * Wo[(4 * k + 3) * OUT_DIM + j];
    }
    out[t] = acc;
  }
}

__global__ void k_mul(const float* __restrict__ a, const float* __restrict__ b,
                      float* __restrict__ out) {
  int t = blockIdx.x * blockDim.x + threadIdx.x;
  if (t < N_NODES * OUT_DIM) out[t] = a[t] * b[t];
}

// ---------------- driver ----------------
extern "C" void kernel_launch(void* const* d_in, const int* in_sizes, int n_in,
                              void* d_out, int out_size, void* d_ws, size_t ws_size,
                              hipStream_t stream) {
  const float* mask_feature = (const float*)d_in[0];
  const float* feature      = (const float*)d_in[1];
  const int*   edge_index   = (const int*)d_in[2];   // [2, E]
  // d_in[3] = edge_type (unused by reference math)
  const float* W_in  = (const float*)d_in[4];
  const float* b_in  = (const float*)d_in[5];
  const float* Wg1   = (const float*)d_in[6];
  const float* bg1   = (const float*)d_in[7];
  const float* Wg2   = (const float*)d_in[8];
  const float* bg2   = (const float*)d_in[9];
  const float* W_out = (const float*)d_in[10];
  const float* b_out = (const float*)d_in[11];
  float* out = (float*)d_out;

  const int* src = edge_index;
  const int* dst = edge_index + N_EDGES;

  // scratch layout (floats)
  float* ws = (float*)d_ws;
  const size_t HSZ = (size_t)N_NODES * H_DIM;     // 12.8M floats
  float* H0  = ws;                                 // hidden buf 0
  float* HA  = H0 + HSZ;                           // hidden buf 1
  float* HB  = HA + HSZ;                           // hidden buf 2
  float* deg = HB + HSZ;
  float* dis = deg + N_NODES;
  float* xo  = dis + N_NODES;                      // main-branch logits [N,2]
  float* mo  = xo + (size_t)N_NODES * OUT_DIM;     // mask-branch logits [N,2]

  const int TB = 256;
  const dim3 b256(TB);

  // 1. degrees & symmetric norm
  k_deg_init<<<dim3((N_NODES + TB - 1) / TB), b256, 0, stream>>>(deg);
  k_deg_count<<<dim3((N_EDGES + TB - 1) / TB), b256, 0, stream>>>(dst, deg);
  k_dis<<<dim3((N_NODES + TB - 1) / TB), b256, 0, stream>>>(deg, dis);

  // GEMM launch geometry: 50000 tiles of 16x16, 4 waves (128 thr) per block
  const int tiles = (N_NODES / 16) * (H_DIM / 16);   // 3125 * 16 = 50000
  const dim3 gemmGrid(tiles / 4), gemmBlk(128);

  // 2. H0 = leaky_relu(mask_feature @ W_in + b_in)
  k_gemm_wmma<E_DIM, true, false><<<gemmGrid, gemmBlk, 0, stream>>>(
      mask_feature, nullptr, W_in, b_in, H0);

  // 3. conv1: HA = H0 @ Wg1 ; HB = aggregate + bg1
  k_gemm_wmma<H_DIM, false, false><<<gemmGrid, gemmBlk, 0, stream>>>(
      H0, nullptr, Wg1, nullptr, HA);
  k_selfinit<<<dim3((N_NODES * H_DIM) / TB), b256, 0, stream>>>(HA, dis, bg1, HB);
  k_scatter<<<dim3((N_EDGES * 64) / TB), b256, 0, stream>>>(src, dst, dis, HA, HB);

  // 4. conv2: H0 = HB @ Wg2 ; HA = aggregate + bg2
  k_gemm_wmma<H_DIM, false, false><<<gemmGrid, gemmBlk, 0, stream>>>(
      HB, nullptr, Wg2, nullptr, H0);
  k_selfinit<<<dim3((N_NODES * H_DIM) / TB), b256, 0, stream>>>(H0, dis, bg2, HA);
  k_scatter<<<dim3((N_EDGES * 64) / TB), b256, 0, stream>>>(src, dst, dis, H0, HA);

  // 5. main-branch logits
  k_outproj<<<dim3((N_NODES * OUT_DIM + TB - 1) / TB), b256, 0, stream>>>(
      HA, W_out, b_out, xo);

  // 6. mask branch: HB = leaky_relu((feature - mask_feature) @ W_in + b_in)
  k_gemm_wmma<E_DIM, true, true><<<gemmGrid, gemmBlk, 0, stream>>>(
      feature, mask_feature, W_in, b_in, HB);
  k_outproj<<<dim3((N_NODES * OUT_DIM + TB - 1) / TB), b256, 0, stream>>>(
      HB, W_out, b_out, mo);

  // 7. out = xo * mo
  k_mul<<<dim3((N_NODES * OUT_DIM + TB - 1) / TB), b256, 0, stream>>>(xo, mo, out);
}